// UniformTorchCorr_15049565405420
// MI455X (gfx1250) — compile-verified
//
#include <hip/hip_runtime.h>
#include <hip/hip_bf16.h>

typedef __attribute__((ext_vector_type(2))) float v2f;
typedef __attribute__((ext_vector_type(8))) float v8f;

#define N_  8
#define C_  128
#define H_  96
#define W_  192
#define R_  4
#define O_  81          // 9x9 offsets
#define TW  16          // output w-tile per workgroup
#define CK  32          // channel chunk staged in LDS
#define NCHUNK (C_ / CK)
#define SRCW 32         // 16-wide tile + 8 halo each side
#define NWAVE 9         // one wave per dy in [-4,4]
#define CP  (CK + 2)    // padded LDS row stride (even -> 8B-aligned pairs)

__global__ __launch_bounds__(NWAVE * 32)
void corr_volume_wmma(const float* __restrict__ fref,
                      const float* __restrict__ fsrc,
                      float* __restrict__ corr,
                      float* __restrict__ mask)
{
    // [m][c] ref tile, transposed so (c,c+1) pairs are contiguous for A frags
    __shared__ float sRef[TW][CP];
    // [dy][col][c] src window per wave, transposed for B frags
    __shared__ float sSrc[NWAVE][SRCW][CP];
    // [dy][m][col] D-tile spill for banded extraction
    __shared__ float sD[NWAVE][TW][SRCW + 2];

    const int t     = threadIdx.x;
    const int wave  = t >> 5;        // dy index 0..8
    const int lane  = t & 31;
    const int laneM = lane & 15;     // M (A) / N (B) index
    const int kHi   = (lane >> 4) << 1;  // lanes 0-15 -> K {0,1}; 16-31 -> K {2,3}

    const int w0 = blockIdx.x * TW;
    const int h  = blockIdx.y;
    const int n  = blockIdx.z;

    const int  dy    = wave - R_;
    const int  srow  = h + dy;
    const bool rowOK = (unsigned)srow < (unsigned)H_;

    const float* refBase = fref + ((size_t)n * C_ * H_ + h) * W_;
    const float* srcBase = fsrc + ((size_t)n * C_ * H_ + (rowOK ? srow : 0)) * W_;

    v8f acc0 = {};   // src columns  0..15 of the 32-wide window
    v8f acc1 = {};   // src columns 16..31

    for (int ch = 0; ch < NCHUNK; ++ch) {
        const int cbase = ch * CK;

        // --- stage ref tile (all 288 threads): coalesced in w, conflict-free in LDS
        for (int i = t; i < CK * TW; i += NWAVE * 32) {
            int cc = i >> 4;        // channel within chunk
            int m  = i & 15;        // pixel within tile
            sRef[m][cc] = refBase[(size_t)(cbase + cc) * (H_ * W_) + (w0 + m)];
        }
        // --- each wave stages its own dy row window (zero-padded)
        for (int i = lane; i < CK * SRCW; i += 32) {
            int cc   = i >> 5;
            int col  = i & 31;
            int wsrc = w0 - 2 * R_ + col;
            float v = 0.0f;
            if (rowOK && (unsigned)wsrc < (unsigned)W_)
                v = srcBase[(size_t)(cbase + cc) * (H_ * W_) + wsrc];
            sSrc[wave][col][cc] = v;
        }
        __syncthreads();

        // --- band GEMM: D(16x16) += A(16x4) x B(4x16), K stepped by 4
        #pragma unroll
        for (int k = 0; k < CK; k += 4) {
            v2f a  = *(const v2f*)&sRef[laneM][k + kHi];
            v2f b0 = *(const v2f*)&sSrc[wave][laneM][k + kHi];
            v2f b1 = *(const v2f*)&sSrc[wave][16 + laneM][k + kHi];
            acc0 = __builtin_amdgcn_wmma_f32_16x16x4_f32(
                       false, a, false, b0, (short)0, acc0, false, false);
            acc1 = __builtin_amdgcn_wmma_f32_16x16x4_f32(
                       false, a, false, b1, (short)0, acc1, false, false);
        }
        __syncthreads();
    }

    // --- spill D tiles to per-wave LDS scratch
    // C/D layout: VGPR r, lanes 0-15 -> M=r, lanes 16-31 -> M=r+8; N = lane&15
    #pragma unroll
    for (int r = 0; r < 8; ++r) {
        int M = r + ((lane >> 4) << 3);
        sD[wave][M][laneM]      = acc0[r];
        sD[wave][M][16 + laneM] = acc1[r];
    }
    // same-wave LDS is in-order; no workgroup barrier needed here

    // --- gather the 9 banded diagonals + mask, write out
    for (int idx = lane; idx < 9 * TW; idx += 32) {
        int dxi = idx >> 4;            // 0..8  (dx = dxi - 4)
        int m   = idx & 15;
        int col = m + R_ + dxi;        // column in 32-wide window
        int w   = w0 + m;
        int o   = wave * 9 + dxi;      // o = (dy+4)*9 + (dx+4)
        size_t oidx = (((size_t)n * O_ + o) * H_ + h) * W_ + w;
        corr[oidx] = sD[wave][m][col];
        mask[oidx] = (rowOK && (unsigned)(w + dxi - R_) < (unsigned)W_) ? 1.0f : 0.0f;
    }
}

extern "C" void kernel_launch(void* const* d_in, const int* in_sizes, int n_in,
                              void* d_out, int out_size, void* d_ws, size_t ws_size,
                              hipStream_t stream) {
    const float* fref = (const float*)d_in[0];
    const float* fsrc = (const float*)d_in[1];
    // d_in[2]: offsets — fixed ±4 meshgrid, baked into the kernel
    float* corr = (float*)d_out;
    float* mask = corr + (size_t)N_ * O_ * H_ * W_;

    dim3 grid(W_ / TW, H_, N_);
    corr_volume_wmma<<<grid, NWAVE * 32, 0, stream>>>(fref, fsrc, corr, mask);
}